// LegalizeCLB_32066225832424
// MI455X (gfx1250) — compile-verified
//
#include <hip/hip_runtime.h>

// -------------------------------------------------------------------------
// Net bbox (HPWL-style) via segmented min/max scatter reduction.
//
// d_out layout (float32, row-major (num_nets,4)): [xl, yl, xh, yh].
// We run u32 bit-pattern atomics directly on d_out:
//   slot0 = atomicMin(x bits)   slot1 = atomicMin(y bits)
//   slot2 = atomicMax(x bits)   slot3 = atomicMax(y bits)
// All coords are >= 0, so unsigned compare of IEEE bits == float compare.
// Accumulators (40MB) are L2-resident on MI455X (192MB L2).
// -------------------------------------------------------------------------

typedef int      v4i __attribute__((ext_vector_type(4)));
typedef float    v4f __attribute__((ext_vector_type(4)));
typedef unsigned v4u __attribute__((ext_vector_type(4)));

#define FLT_MAX_BITS 0x7F7FFFFFu

__global__ void bbox_init_kernel(unsigned* __restrict__ out, int num_nets) {
    int i = blockIdx.x * blockDim.x + threadIdx.x;
    if (i < num_nets) {
        v4u v = {FLT_MAX_BITS, FLT_MAX_BITS, 0u, 0u}; // min,min,max,max identities
        reinterpret_cast<v4u*>(out)[i] = v;
    }
}

__device__ __forceinline__ void scatter_pin(const float* __restrict__ pos_x,
                                            const float* __restrict__ pos_y,
                                            unsigned* __restrict__ out,
                                            int node, int net,
                                            float ox, float oy) {
    // pos gathers: default (RT) temporal hint -> cached; pos arrays (16MB)
    // stay hot in L2 alongside the 40MB atomic destination region.
    float px = pos_x[node] + ox;
    float py = pos_y[node] + oy;
    unsigned xb = __float_as_uint(px);
    unsigned yb = __float_as_uint(py);
    unsigned* row = out + ((long long)net << 2);
    atomicMin(row + 0, xb);   // GLOBAL_ATOMIC_MIN_U32 (no return -> STOREcnt)
    atomicMin(row + 1, yb);
    atomicMax(row + 2, xb);   // GLOBAL_ATOMIC_MAX_U32
    atomicMax(row + 3, yb);
}

__global__ void bbox_pin_kernel(const float* __restrict__ pos_x,
                                const float* __restrict__ pos_y,
                                const int*   __restrict__ pin2node,
                                const int*   __restrict__ pin2net,
                                const float* __restrict__ offx,
                                const float* __restrict__ offy,
                                unsigned*    __restrict__ out,
                                int num_pins) {
    int tid  = blockIdx.x * blockDim.x + threadIdx.x;
    int base = tid * 4;
    if (base + 3 < num_pins) {
        // Streaming, one-shot data: non-temporal b128 loads so 256MB of pin
        // streams don't evict the L2-hot pos/accumulator working set.
        v4i nodes = __builtin_nontemporal_load(
                        reinterpret_cast<const v4i*>(pin2node + base));
        v4i nets  = __builtin_nontemporal_load(
                        reinterpret_cast<const v4i*>(pin2net + base));
        v4f ox    = __builtin_nontemporal_load(
                        reinterpret_cast<const v4f*>(offx + base));
        v4f oy    = __builtin_nontemporal_load(
                        reinterpret_cast<const v4f*>(offy + base));
        scatter_pin(pos_x, pos_y, out, nodes.x, nets.x, ox.x, oy.x);
        scatter_pin(pos_x, pos_y, out, nodes.y, nets.y, ox.y, oy.y);
        scatter_pin(pos_x, pos_y, out, nodes.z, nets.z, ox.z, oy.z);
        scatter_pin(pos_x, pos_y, out, nodes.w, nets.w, ox.w, oy.w);
    } else {
        for (int i = base; i < num_pins; ++i) {
            scatter_pin(pos_x, pos_y, out, pin2node[i], pin2net[i],
                        offx[i], offy[i]);
        }
    }
}

__global__ void bbox_finalize_kernel(unsigned* __restrict__ out, int num_nets) {
    int i = blockIdx.x * blockDim.x + threadIdx.x;
    if (i < num_nets) {
        v4u v = reinterpret_cast<const v4u*>(out)[i];
        // Empty net <=> min slot untouched (no coordinate can equal FLT_MAX).
        if (v.x == FLT_MAX_BITS) {
            v4u z = {0u, 0u, 0u, 0u};
            reinterpret_cast<v4u*>(out)[i] = z;
        }
        // Valid nets: bits already ARE the float bbox values; no write needed.
    }
}

extern "C" void kernel_launch(void* const* d_in, const int* in_sizes, int n_in,
                              void* d_out, int out_size, void* d_ws, size_t ws_size,
                              hipStream_t stream) {
    const float* pos      = (const float*)d_in[0];   // 2*n floats: x then y
    const int*   pin2node = (const int*)  d_in[1];
    const int*   pin2net  = (const int*)  d_in[2];
    const float* offx     = (const float*)d_in[3];
    const float* offy     = (const float*)d_in[4];
    // d_in[5] = net_wts (unused by the reference output)

    const int num_nodes = in_sizes[0] / 2;
    const int num_pins  = in_sizes[1];
    const int num_nets  = in_sizes[5];

    const float* pos_x = pos;
    const float* pos_y = pos + num_nodes;

    unsigned* out_u = (unsigned*)d_out;

    const int BLK = 256;  // 8 wave32s per block

    // 1) init accumulators in d_out (also clears harness poison)
    int init_blocks = (num_nets + BLK - 1) / BLK;
    bbox_init_kernel<<<init_blocks, BLK, 0, stream>>>(out_u, num_nets);

    // 2) scatter atomics, 4 pins per thread
    int nthreads   = (num_pins + 3) / 4;
    int pin_blocks = (nthreads + BLK - 1) / BLK;
    bbox_pin_kernel<<<pin_blocks, BLK, 0, stream>>>(
        pos_x, pos_y, pin2node, pin2net, offx, offy, out_u, num_pins);

    // 3) zero out empty nets
    bbox_finalize_kernel<<<init_blocks, BLK, 0, stream>>>(out_u, num_nets);
}